// PPOContinuous_16527034155702
// MI455X (gfx1250) — compile-verified
//
#include <hip/hip_runtime.h>
#include <cstdint>
#include <cstddef>

#ifndef __has_builtin
#define __has_builtin(x) 0
#endif

#if defined(__gfx1250__) && __has_builtin(__builtin_amdgcn_tensor_load_to_lds)
#define USE_TDM 1
#else
#define USE_TDM 0
#endif

#if defined(__gfx1250__) && __has_builtin(__builtin_amdgcn_wmma_f32_16x16x4_f32)
#define USE_WMMA 1
#else
#define USE_WMMA 0
#endif

#if defined(__gfx1250__) && __has_builtin(__builtin_amdgcn_s_wait_tensorcnt)
#define WAIT_TENSORCNT(n) __builtin_amdgcn_s_wait_tensorcnt((short)(n))
#else
#define WAIT_TENSORCNT(n) do {} while (0)
#endif

// ---- geometry -------------------------------------------------------------
#define NCOL            64          // action dimension N
#define TILE_ROWS       64          // rows per block-tile
#define ROWS_PER_WAVE   16          // rows handled by one wave32
#define WAVES_PER_BLK   4
#define BLK_THREADS     128
#define ROW_STRIDE_DW   68          // 64 data dwords + 4 pad dwords (TDM pad feature)
#define ARR_BYTES       (TILE_ROWS * ROW_STRIDE_DW * 4)   // 17408 B per array per buffer
#define BUF_BYTES       (5 * ARR_BYTES)                   // 87040 B per buffer
#define ROWSCR_OFF      (2 * BUF_BYTES)                   // 174080
#define ROWSCR_BYTES    (WAVES_PER_BLK * 48 * 4)          // 3 matrices x 16 rows per wave
#define RED_OFF         (ROWSCR_OFF + ROWSCR_BYTES)       // 174848 (8B aligned)
#define SMEM_BYTES      (RED_OFF + 5 * 8)                 // + 5 doubles
#define GRID_BLOCKS     1024

#define HALF_LOG_2PI_F  0.9189385332046727f
#define LN2_F           0.6931471805599453f
#define LOG2E_F         1.4426950408889634f

typedef __attribute__((ext_vector_type(2))) float v2f;
typedef __attribute__((ext_vector_type(8))) float v8f;
typedef __attribute__((ext_vector_type(4))) unsigned tdm_u4;
typedef __attribute__((ext_vector_type(8))) int      tdm_i8;
typedef __attribute__((ext_vector_type(4))) int      tdm_i4;

// ---- fast VALU transcendentals (inputs are well inside hardware range) ----
__device__ __forceinline__ float fast_log(float x) {
#if defined(__gfx1250__) && __has_builtin(__builtin_amdgcn_logf)
    return __builtin_amdgcn_logf(x) * LN2_F;      // v_log_f32 (log2) * ln2
#else
    return __logf(x);
#endif
}
__device__ __forceinline__ float fast_exp(float x) {
#if defined(__gfx1250__) && __has_builtin(__builtin_amdgcn_exp2f)
    return __builtin_amdgcn_exp2f(x * LOG2E_F);   // v_exp_f32 (2^x)
#else
    return __expf(x);
#endif
}
__device__ __forceinline__ float fast_rcp(float x) {
#if defined(__gfx1250__) && __has_builtin(__builtin_amdgcn_rcpf)
    return __builtin_amdgcn_rcpf(x);              // v_rcp_f32, 1 ULP
#else
    return 1.0f / x;
#endif
}

// ---- TDM: issue one 64x64 f32 tile -> LDS (rows padded to 68 dwords) ------
#if USE_TDM
__device__ __forceinline__ void tdm_tile_to_lds(const float* gtile,
                                                unsigned lds_byte_addr,
                                                int total_rows)
{
    const unsigned long long ga = (unsigned long long)(uintptr_t)gtile;

    tdm_u4 g0;
    g0[0] = 1u;                                       // count=1, user descriptor
    g0[1] = lds_byte_addr;                            // LDS destination (bytes)
    g0[2] = (unsigned)(ga & 0xFFFFFFFFull);           // global_addr[31:0]
    g0[3] = (unsigned)((ga >> 32) & 0x1FFFFFFull)     // global_addr[56:32]
          | (2u << 30);                               // type = 2 ("image")

    tdm_i8 g1;
    // data_size=2 (4B) | pad_enable | pad_interval=5 (64 dwords) | pad_amount=3 (4 dwords)
    g1[0] = (int)((2u << 16) | (1u << 20) | (5u << 22) | (3u << 25));
    g1[1] = (int)(64u << 16);                         // tensor_dim0[15:0]=64 in [63:48]
    g1[2] = (int)(((unsigned)total_rows & 0xFFFFu) << 16);  // tensor_dim1[15:0]
    g1[3] = (int)(((unsigned)total_rows >> 16) | (64u << 16)); // tensor_dim1 hi | tile_dim0=64
    g1[4] = (int)TILE_ROWS;                           // tile_dim1=64, tile_dim2=0
    g1[5] = 64;                                       // tensor_dim0_stride = 64 elements
    g1[6] = 0;
    g1[7] = 0;

    tdm_i4 gz4; gz4[0] = 0; gz4[1] = 0; gz4[2] = 0; gz4[3] = 0;   // groups 2/3 unused (2-D)
    tdm_i8 gz8;
    gz8[0] = 0; gz8[1] = 0; gz8[2] = 0; gz8[3] = 0;
    gz8[4] = 0; gz8[5] = 0; gz8[6] = 0; gz8[7] = 0;

    // clang-23 / therock-10.0 form: (u32x4, i32x8, i32x4, i32x4, i32x8, i32 cpol)
    __builtin_amdgcn_tensor_load_to_lds(g0, g1, gz4, gz4, gz8, 0);
}
#else
__device__ __forceinline__ void manual_tile_to_lds(const float* gtile, float* lbase)
{
    for (int i = threadIdx.x; i < TILE_ROWS * (NCOL / 4); i += BLK_THREADS) {
        const int r  = i >> 4;
        const int c4 = (i & 15) << 2;
        const float4 v = reinterpret_cast<const float4*>(gtile)[i];
        *reinterpret_cast<float4*>(lbase + r * ROW_STRIDE_DW + c4) = v;
    }
}
#endif

#if USE_WMMA
__device__ __forceinline__ v8f wmma_rowsum_step(float a_lo, float a_hi, v8f c)
{
    v2f a; a[0] = a_lo; a[1] = a_hi;      // A: 16x4 f32 slice (this lane's 2 K-values)
    v2f b; b[0] = 1.0f; b[1] = 1.0f;      // B: 4x16 all-ones -> D[m][n] = sum_k A[m][k] + C
    return __builtin_amdgcn_wmma_f32_16x16x4_f32(false, a, false, b,
                                                 (short)0, c, false, false);
}
#endif

// process 4 consecutive elements of one row
__device__ __forceinline__ void quad_partials(const float4 mn, const float4 sn,
                                              const float4 mo, const float4 so,
                                              const float4 ac,
                                              float& q1, float& q2, float& q3)
{
    float s1 = 0.f, s2 = 0.f, s3 = 0.f;
#define PPO_ELEM(c)                                                        \
    {                                                                      \
        const float ln = fast_log(sn.c);                                   \
        const float lo = fast_log(so.c);                                   \
        const float zn = (ac.c - mn.c) * fast_rcp(sn.c);                   \
        const float zo = (ac.c - mo.c) * fast_rcp(so.c);                   \
        s1 += -0.5f * zn * zn - ln;                                        \
        s2 += -0.5f * zo * zo - lo;                                        \
        s3 += ln;                                                          \
    }
    PPO_ELEM(x) PPO_ELEM(y) PPO_ELEM(z) PPO_ELEM(w)
#undef PPO_ELEM
    q1 = s1; q2 = s2; q3 = s3;
}

__global__ void __launch_bounds__(BLK_THREADS)
ppo_main_kernel(const float* __restrict__ mu_new, const float* __restrict__ sg_new,
                const float* __restrict__ mu_old, const float* __restrict__ sg_old,
                const float* __restrict__ act,
                const float* __restrict__ v_new,  const float* __restrict__ v_old,
                const float* __restrict__ adv,    const float* __restrict__ ret,
                const float* __restrict__ wgt,
                double* __restrict__ gacc, int Brows)
{
    extern __shared__ char smem[];
    const int tid  = threadIdx.x;
    const int w    = tid >> 5;       // wave id in block (wave32)
    const int lane = tid & 31;
    const int m    = lane & 15;      // WMMA A-matrix row index
    const int h    = lane >> 4;      // half selector (K pair {0,1} vs {2,3})
    const int T    = Brows / TILE_ROWS;
    const int G    = gridDim.x;

    double acc_pol = 0.0, acc_val = 0.0, acc_ent = 0.0, acc_kl = 0.0, acc_cf = 0.0;

    const int t0 = (int)blockIdx.x;

#if USE_TDM
    if (w == 0 && t0 < T) {
        const size_t off = (size_t)t0 * TILE_ROWS * NCOL;
        tdm_tile_to_lds(mu_new + off, (unsigned)(uintptr_t)(smem + 0 * ARR_BYTES), Brows);
        tdm_tile_to_lds(sg_new + off, (unsigned)(uintptr_t)(smem + 1 * ARR_BYTES), Brows);
        tdm_tile_to_lds(mu_old + off, (unsigned)(uintptr_t)(smem + 2 * ARR_BYTES), Brows);
        tdm_tile_to_lds(sg_old + off, (unsigned)(uintptr_t)(smem + 3 * ARR_BYTES), Brows);
        tdm_tile_to_lds(act    + off, (unsigned)(uintptr_t)(smem + 4 * ARR_BYTES), Brows);
    }
#endif

    int buf = 0;
    for (int t = t0; t < T; t += G) {
        const int tn = t + G;

#if USE_TDM
        if (w == 0) {
            if (tn < T) {   // prefetch next tile into the other buffer, then wait for current
                const size_t off = (size_t)tn * TILE_ROWS * NCOL;
                char* nb = smem + (buf ^ 1) * BUF_BYTES;
                tdm_tile_to_lds(mu_new + off, (unsigned)(uintptr_t)(nb + 0 * ARR_BYTES), Brows);
                tdm_tile_to_lds(sg_new + off, (unsigned)(uintptr_t)(nb + 1 * ARR_BYTES), Brows);
                tdm_tile_to_lds(mu_old + off, (unsigned)(uintptr_t)(nb + 2 * ARR_BYTES), Brows);
                tdm_tile_to_lds(sg_old + off, (unsigned)(uintptr_t)(nb + 3 * ARR_BYTES), Brows);
                tdm_tile_to_lds(act    + off, (unsigned)(uintptr_t)(nb + 4 * ARR_BYTES), Brows);
                WAIT_TENSORCNT(5);   // wait for current tile's 5 loads; next 5 stay in flight
            } else {
                WAIT_TENSORCNT(0);
            }
        }
#else
        {
            const size_t off = (size_t)t * TILE_ROWS * NCOL;
            char* cb = smem + buf * BUF_BYTES;
            manual_tile_to_lds(mu_new + off, (float*)(cb + 0 * ARR_BYTES));
            manual_tile_to_lds(sg_new + off, (float*)(cb + 1 * ARR_BYTES));
            manual_tile_to_lds(mu_old + off, (float*)(cb + 2 * ARR_BYTES));
            manual_tile_to_lds(sg_old + off, (float*)(cb + 3 * ARR_BYTES));
            manual_tile_to_lds(act    + off, (float*)(cb + 4 * ARR_BYTES));
        }
#endif
        __syncthreads();   // tile data visible to all waves

        // ---- elementwise Gaussian log-prob partials, accumulated directly in
        // ---- V_WMMA_F32_16X16X4_F32 A-matrix layout: lane l owns row m=l%16,
        // ---- K slots {2h, 2h+1} of slice j  <=>  columns [16j+8h, 16j+8h+8)
        const int lrow = w * ROWS_PER_WAVE + m;
        const char* cb = smem + buf * BUF_BYTES;
        const float* Lmn = (const float*)(cb + 0 * ARR_BYTES) + lrow * ROW_STRIDE_DW;
        const float* Lsn = (const float*)(cb + 1 * ARR_BYTES) + lrow * ROW_STRIDE_DW;
        const float* Lmo = (const float*)(cb + 2 * ARR_BYTES) + lrow * ROW_STRIDE_DW;
        const float* Lso = (const float*)(cb + 3 * ARR_BYTES) + lrow * ROW_STRIDE_DW;
        const float* Lac = (const float*)(cb + 4 * ARR_BYTES) + lrow * ROW_STRIDE_DW;

        float a1[4][2], a2[4][2], a3[4][2];
#pragma unroll
        for (int j = 0; j < 4; ++j) {
            const int c0 = 16 * j + 8 * h;
            const float4 mn0 = *(const float4*)(Lmn + c0);
            const float4 mn1 = *(const float4*)(Lmn + c0 + 4);
            const float4 sn0 = *(const float4*)(Lsn + c0);
            const float4 sn1 = *(const float4*)(Lsn + c0 + 4);
            const float4 mo0 = *(const float4*)(Lmo + c0);
            const float4 mo1 = *(const float4*)(Lmo + c0 + 4);
            const float4 so0 = *(const float4*)(Lso + c0);
            const float4 so1 = *(const float4*)(Lso + c0 + 4);
            const float4 ac0 = *(const float4*)(Lac + c0);
            const float4 ac1 = *(const float4*)(Lac + c0 + 4);
            quad_partials(mn0, sn0, mo0, so0, ac0, a1[j][0], a2[j][0], a3[j][0]);
            quad_partials(mn1, sn1, mo1, so1, ac1, a1[j][1], a2[j][1], a3[j][1]);
        }

        float s1, s2, s3;                     // per-row sums (valid on lanes 0..15)
#if USE_WMMA
        v8f d1 = {0,0,0,0,0,0,0,0};
        v8f d2 = {0,0,0,0,0,0,0,0};
        v8f d3 = {0,0,0,0,0,0,0,0};
#pragma unroll
        for (int j = 0; j < 4; ++j) {
            d1 = wmma_rowsum_step(a1[j][0], a1[j][1], d1);
            d2 = wmma_rowsum_step(a2[j][0], a2[j][1], d2);
            d3 = wmma_rowsum_step(a3[j][0], a3[j][1], d3);
        }
        // D layout: lanes 0-15 vgpr v = row v; lanes 16-31 vgpr v = row v+8
        float* rsc = (float*)(smem + ROWSCR_OFF) + w * 48;
        if (lane == 0) {
#pragma unroll
            for (int v = 0; v < 8; ++v) { rsc[v] = d1[v]; rsc[16 + v] = d2[v]; rsc[32 + v] = d3[v]; }
        } else if (lane == 16) {
#pragma unroll
            for (int v = 0; v < 8; ++v) { rsc[8 + v] = d1[v]; rsc[24 + v] = d2[v]; rsc[40 + v] = d3[v]; }
        }
        __syncthreads();
        s1 = rsc[m]; s2 = rsc[16 + m]; s3 = rsc[32 + m];
#else
        s1 = 0.f; s2 = 0.f; s3 = 0.f;
#pragma unroll
        for (int j = 0; j < 4; ++j) {
            s1 += a1[j][0] + a1[j][1];
            s2 += a2[j][0] + a2[j][1];
            s3 += a3[j][0] + a3[j][1];
        }
        s1 += __shfl_xor(s1, 16, 32);
        s2 += __shfl_xor(s2, 16, 32);
        s3 += __shfl_xor(s3, 16, 32);
#endif

        if (lane < 16) {   // one row per lane: PPO scalar math
            const int   grow  = t * TILE_ROWS + w * ROWS_PER_WAVE + lane;
            const float advv  = adv[grow];
            const float ww    = wgt[grow];
            const float vn    = v_new[grow];
            const float vo    = v_old[grow];
            const float rt    = ret[grow];

            const float delta = s1 - s2;                 // logp_new - logp_old (consts cancel)
            const float ratio = fast_exp(delta);
            const float surr1 = ratio * advv;
            const float rclip = fminf(fmaxf(ratio, 0.8f), 1.2f);
            const float surr2 = rclip * advv;
            float clipped = fminf(surr1, surr2);
            clipped = fmaxf(clipped, 5.0f * advv);       // dual clip

            const float entropy = 64.0f * (0.5f + HALF_LOG_2PI_F) + s3;

            const float vclip = vo + fminf(fmaxf(vn - vo, -0.2f), 0.2f);
            const float e1 = rt - vn,    e2 = rt - vclip;
            const float vmax = fmaxf(e1 * e1, e2 * e2);

            acc_pol += (double)(clipped * ww);
            acc_val += (double)(vmax * ww);
            acc_ent += (double)(entropy * ww);
            acc_kl  += (double)(s2 - s1);                // logp_old - logp_new
            acc_cf  += (fabsf(ratio - 1.0f) > 0.2f) ? 1.0 : 0.0;
        }
        __syncthreads();   // everyone done with buf before its next DMA refill
        buf ^= 1;
    }

    // ---- block reduction: LDS f64 atomics, then one global f64 atomic -----
    double* red = (double*)(smem + RED_OFF);
    if (tid < 5) red[tid] = 0.0;
    __syncthreads();
    if (lane < 16) {
        atomicAdd(&red[0], acc_pol);
        atomicAdd(&red[1], acc_val);
        atomicAdd(&red[2], acc_ent);
        atomicAdd(&red[3], acc_kl);
        atomicAdd(&red[4], acc_cf);
    }
    __syncthreads();
    if (tid == 0) {
        atomicAdd(&gacc[0], red[0]);
        atomicAdd(&gacc[1], red[1]);
        atomicAdd(&gacc[2], red[2]);
        atomicAdd(&gacc[3], red[3]);
        atomicAdd(&gacc[4], red[4]);
    }
}

__global__ void ppo_finalize_kernel(const double* __restrict__ gacc,
                                    float* __restrict__ out, int Brows)
{
    if (threadIdx.x == 0 && blockIdx.x == 0) {
        const double invB = 1.0 / (double)Brows;
        out[0] = (float)(-gacc[0] * invB);        // policy_loss
        out[1] = (float)(0.5 * gacc[1] * invB);   // value_loss
        out[2] = (float)(gacc[2] * invB);         // entropy_loss
        out[3] = (float)(gacc[3] * invB);         // approx_kl
        out[4] = (float)(gacc[4] * invB);         // clipfrac
    }
}

extern "C" void kernel_launch(void* const* d_in, const int* in_sizes, int n_in,
                              void* d_out, int out_size, void* d_ws, size_t ws_size,
                              hipStream_t stream)
{
    const float* mu_new = (const float*)d_in[0];
    const float* sg_new = (const float*)d_in[1];
    const float* mu_old = (const float*)d_in[2];
    const float* sg_old = (const float*)d_in[3];
    const float* act    = (const float*)d_in[4];
    const float* v_new  = (const float*)d_in[5];
    const float* v_old  = (const float*)d_in[6];
    const float* adv    = (const float*)d_in[7];
    const float* ret    = (const float*)d_in[8];
    const float* wgt    = (const float*)d_in[9];
    const int Brows     = in_sizes[5];            // (B,) array length

    double* gacc = (double*)d_ws;
    (void)ws_size; (void)n_in; (void)out_size;

    hipError_t e0 = hipFuncSetAttribute((const void*)ppo_main_kernel,
                                        hipFuncAttributeMaxDynamicSharedMemorySize,
                                        SMEM_BYTES);
    (void)e0;

    hipError_t e1 = hipMemsetAsync(gacc, 0, 5 * sizeof(double), stream);
    (void)e1;

    dim3 grid(GRID_BLOCKS), block(BLK_THREADS);
    ppo_main_kernel<<<grid, block, SMEM_BYTES, stream>>>(
        mu_new, sg_new, mu_old, sg_old, act, v_new, v_old, adv, ret, wgt,
        gacc, Brows);

    ppo_finalize_kernel<<<1, 32, 0, stream>>>(gacc, (float*)d_out, Brows);
}